// VectorQuantizer_8641474200352
// MI455X (gfx1250) — compile-verified
//
#include <hip/hip_runtime.h>

typedef __attribute__((ext_vector_type(2))) float v2f;
typedef __attribute__((ext_vector_type(4))) float v4f;
typedef __attribute__((ext_vector_type(8))) float v8f;
typedef __attribute__((ext_vector_type(4))) unsigned int v4u;
typedef __attribute__((ext_vector_type(4))) int v4i;
typedef __attribute__((ext_vector_type(8))) int v8i;

#define N_EMB 1024
#define DIM 64
#define PAD 68                     // padded LDS row stride (floats): 16B aligned, spreads banks
#define WAVES 8
#define ROWS_PER_WAVE 32           // two 16-row WMMA tiles per wave
#define ROWS_PER_WG (WAVES * ROWS_PER_WAVE)   // 256
#define NROWS (128 * 32 * 32)      // 131072
#define QUANT_ELEMS ((size_t)NROWS * DIM)

__global__ __launch_bounds__(256) void vq_wmma_kernel(
    const float* __restrict__ X,      // [NROWS, 64]
    const float* __restrict__ E,      // [1024, 64]
    float* __restrict__ quant,        // [NROWS, 64]
    float* __restrict__ enc)          // [NROWS, 1024]
{
    extern __shared__ float smem[];
    float* cb    = smem;                    // [1024][PAD]  (PAD via TDM lds padding)
    float* enorm = smem + N_EMB * PAD;      // [1024]

    const int tid  = threadIdx.x;
    const int lane = tid & 31;
    const int wave = tid >> 5;
    const int half = lane >> 4;    // 0: K=0,1 ; 1: K=2,3
    const int l15  = lane & 15;
    const int koff = half * 2;

    // ---- stage codebook into LDS via Tensor Data Mover (async DMA) ----
    // 2D tensor 1024x64 f32, tile = whole tensor; TDM pads LDS dest by 4 DWORDs
    // every 64 DWORDs -> effective row stride = 68 floats (bank-conflict-free).
    if (wave == 0) {
        unsigned long long ga = (unsigned long long)(uintptr_t)E;
        unsigned int ldsa = (unsigned int)(uintptr_t)cb;   // low 32b of flat = LDS byte offset
        v4u g0;
        g0.x = 1u;                                          // count=1 (valid descriptor)
        g0.y = ldsa;                                        // lds_addr
        g0.z = (unsigned int)(ga & 0xFFFFFFFFu);            // global_addr[31:0]
        g0.w = (unsigned int)((ga >> 32) & 0x01FFFFFFu)     // global_addr[56:32]
             | 0x80000000u;                                 // type=2 ("image")
        v8i g1;
        g1[0] = (2 << 16)      // data_size = 4 bytes
              | (1 << 20)      // pad_enable
              | (5 << 22)      // pad_interval: every 64 DWORDs (one 64-float row)
              | (3 << 25);     // pad_amount: 4 DWORDs -> 68-float LDS stride
        g1[1] = (int)(64u << 16);     // tensor_dim0 = 64   (bits 79:48, low half)
        g1[2] = (int)(1024u << 16);   // tensor_dim1 = 1024 (bits 95:80)
        g1[3] = (int)(64u << 16);     // tensor_dim1 hi=0, tile_dim0 = 64 (bits 127:112)
        g1[4] = 1024;                 // tile_dim1 = 1024, tile_dim2 = 0
        g1[5] = 64;                   // tensor_dim0_stride = 64
        g1[6] = 0;
        g1[7] = 0;
        v4i z4 = {0, 0, 0, 0};
        v8i z8 = {0, 0, 0, 0, 0, 0, 0, 0};
        __builtin_amdgcn_tensor_load_to_lds(g0, g1, z4, z4, z8, 0);
    }

    // ---- overlap: load A fragments for 2 row-tiles while TDM runs ----
    const int rowBase = blockIdx.x * ROWS_PER_WG + wave * ROWS_PER_WAVE;
    v2f a[2][16];
    #pragma unroll
    for (int r = 0; r < 2; ++r) {
        const float* xrow = X + (size_t)(rowBase + r * 16 + l15) * DIM;
        #pragma unroll
        for (int f = 0; f < 16; ++f)
            a[r][f] = *(const v2f*)(xrow + f * 4 + koff);
    }

    if (wave == 0) __builtin_amdgcn_s_wait_tensorcnt(0);
    __syncthreads();

    // ---- per-code squared norms ----
    for (int c = tid; c < N_EMB; c += 256) {
        float s = 0.f;
        const float* r = cb + c * PAD;
        #pragma unroll 8
        for (int d = 0; d < DIM; ++d) { float e = r[d]; s += e * e; }
        enorm[c] = s;
    }
    __syncthreads();

    float bestS[2][8];
    int   bestI[2][8];
    #pragma unroll
    for (int r = 0; r < 2; ++r)
        #pragma unroll
        for (int v = 0; v < 8; ++v) { bestS[r][v] = 3.4e38f; bestI[r][v] = 0; }

    // ---- sweep 64 code-tiles; S = X · Eᵀ via chained f32 WMMA ----
    for (int t = 0; t < N_EMB / 16; ++t) {
        const int   code = t * 16 + l15;            // this lane's candidate code (column)
        const float en   = enorm[code];
        const float* brow = cb + code * PAD + koff; // E[code, koff..]
        v8f acc0 = {}; v8f acc1 = {};
        #pragma unroll
        for (int f = 0; f < 16; ++f) {
            v2f b = *(const v2f*)(brow + f * 4);    // B[k][n] = E[code n][k]
            acc0 = __builtin_amdgcn_wmma_f32_16x16x4_f32(
                       false, a[0][f], false, b, (short)0, acc0, false, false);
            acc1 = __builtin_amdgcn_wmma_f32_16x16x4_f32(
                       false, a[1][f], false, b, (short)0, acc1, false, false);
        }
        #pragma unroll
        for (int v = 0; v < 8; ++v) {
            float s0 = en - 2.0f * acc0[v];
            float s1 = en - 2.0f * acc1[v];
            bool c0 = s0 < bestS[0][v];
            bool c1 = s1 < bestS[1][v];
            bestS[0][v] = c0 ? s0 : bestS[0][v];
            bestI[0][v] = c0 ? code : bestI[0][v];
            bestS[1][v] = c1 ? s1 : bestS[1][v];
            bestI[1][v] = c1 ? code : bestI[1][v];
        }
    }

    // ---- cross-lane argmin: butterfly over the 16 lanes sharing each row ----
    #pragma unroll
    for (int r = 0; r < 2; ++r) {
        #pragma unroll
        for (int v = 0; v < 8; ++v) {
            float s = bestS[r][v]; int i = bestI[r][v];
            #pragma unroll
            for (int m = 1; m <= 8; m <<= 1) {
                float os = __shfl_xor(s, m, 32);
                int   oi = __shfl_xor(i, m, 32);
                bool take = (os < s) || (os == s && oi < i);  // first-occurrence tie-break
                s = take ? os : s;
                i = take ? oi : i;
            }
            bestI[r][v] = i;
        }
    }
    // broadcast per-row winners to all lanes (rows 0-7 live in lanes 0-15, 8-15 in 16-31)
    int idxAll[2][16];
    #pragma unroll
    for (int r = 0; r < 2; ++r)
        #pragma unroll
        for (int v = 0; v < 16; ++v)
            idxAll[r][v] = __shfl(bestI[r][v & 7], (v < 8) ? 0 : 16, 32);

    // ---- write quantized (gather from LDS codebook) + one-hot encodings ----
    #pragma unroll
    for (int r = 0; r < 2; ++r) {
        #pragma unroll
        for (int v = 0; v < 16; ++v) {
            const int row  = rowBase + r * 16 + v;
            const int code = idxAll[r][v];
            // quantized row: 32 lanes x 2 floats, conflict-free ds_load_b64
            v2f q = *(const v2f*)(cb + code * PAD + lane * 2);
            *(v2f*)(quant + (size_t)row * DIM + lane * 2) = q;
            // encodings row: lane owns contiguous [lane*32, lane*32+32)
            float* erow = enc + (size_t)row * N_EMB + lane * 32;
            #pragma unroll
            for (int j = 0; j < 8; ++j) {
                int cbase = lane * 32 + j * 4;
                v4f val;
                val.x = (cbase + 0 == code) ? 1.0f : 0.0f;
                val.y = (cbase + 1 == code) ? 1.0f : 0.0f;
                val.z = (cbase + 2 == code) ? 1.0f : 0.0f;
                val.w = (cbase + 3 == code) ? 1.0f : 0.0f;
                __builtin_nontemporal_store(val, (v4f*)(erow + j * 4));
            }
        }
    }
}

extern "C" void kernel_launch(void* const* d_in, const int* in_sizes, int n_in,
                              void* d_out, int out_size, void* d_ws, size_t ws_size,
                              hipStream_t stream) {
    (void)in_sizes; (void)n_in; (void)d_ws; (void)ws_size; (void)out_size;
    const float* X = (const float*)d_in[0];   // inputs  [128,32,32,64] fp32
    const float* E = (const float*)d_in[1];   // codebook [1024,64] fp32
    float* quant = (float*)d_out;             // first output, 8388608 floats
    float* enc   = quant + QUANT_ELEMS;       // second output, 134217728 floats

    const size_t lds_bytes = (size_t)(N_EMB * PAD + N_EMB) * sizeof(float); // ~276 KB
    (void)hipFuncSetAttribute(reinterpret_cast<const void*>(vq_wmma_kernel),
                              hipFuncAttributeMaxDynamicSharedMemorySize, (int)lds_bytes);

    dim3 grid(NROWS / ROWS_PER_WG);   // 512 workgroups
    dim3 block(256);                  // 8 wave32
    vq_wmma_kernel<<<grid, block, lds_bytes, stream>>>(X, E, quant, enc);
}